// ScaledDotProductAttention_60533269070053
// MI455X (gfx1250) — compile-verified
//
#include <hip/hip_runtime.h>

typedef __attribute__((ext_vector_type(2))) float v2f;
typedef __attribute__((ext_vector_type(4))) float v4f;
typedef __attribute__((ext_vector_type(8))) float v8f;

#define B_ 4
#define H_ 8
#define S_ 2048
#define D_ 16
#define NT (S_ / 16)          // 128 key tiles per row block
#define WAVES_PER_BLOCK 8
#define PT_STRIDE 20          // padded LDS tile row stride (floats)

__launch_bounds__(WAVES_PER_BLOCK * 32)
__global__ void sdpa_wmma_f32_kernel(const float* __restrict__ Q,
                                     const float* __restrict__ K,
                                     const float* __restrict__ V,
                                     const int*   __restrict__ Msk,
                                     float* __restrict__ out)
{
    __shared__ float    ptile[WAVES_PER_BLOCK][16 * PT_STRIDE]; // 10 KB
    __shared__ unsigned mlds [WAVES_PER_BLOCK][NT / 4][32];     // 32 KB: packed mask bits

    const int tid  = threadIdx.x;
    const int lane = tid & 31;
    const int wave = tid >> 5;
    const int half = lane >> 4;     // 0: lanes 0-15, 1: lanes 16-31
    const int l15  = lane & 15;
    const int koff = half << 1;     // K offset within a 4-chunk: 0 or 2

    const int gw = blockIdx.x * WAVES_PER_BLOCK + wave;  // 0..4095
    const int qb = gw & (NT - 1);                        // q-block within head
    const int bh = gw >> 7;                              // b*H + h (same for whole block)
    const int b  = bh >> 3;                              // H_ == 8
    const int q0 = qb << 4;

    const int qkvBase = bh * (S_ * D_);

    // ---- Q A-fragments (16x4 f32 per chunk), live for both passes ----
    v2f qa[4];
    {
        const float* qrow = Q + qkvBase + (q0 + l15) * D_;
        #pragma unroll
        for (int j = 0; j < 4; ++j)
            qa[j] = *(const v2f*)(qrow + j * 4 + koff);
    }

    // ---------------- pass 1: online row-max / row-sum + mask-bit capture ----
    float rm[8], rs[8];
    #pragma unroll
    for (int r = 0; r < 8; ++r) { rm[r] = -3.0e38f; rs[r] = 0.0f; }

    unsigned mb = 0;
    for (int kt = 0; kt < NT; ++kt) {
        const int key0 = kt << 4;
        const float* krow = K + qkvBase + (key0 + l15) * D_;

        v8f acc = {};
        #pragma unroll
        for (int j = 0; j < 4; ++j) {
            v2f kb = *(const v2f*)(krow + j * 4 + koff);
            acc = __builtin_amdgcn_wmma_f32_16x16x4_f32(false, qa[j], false, kb,
                                                        (short)0, acc, false, false);
        }

        if (kt + 1 < NT) {
            __builtin_prefetch(K + qkvBase + (key0 + 16 + l15) * D_, 0, 1);
            __builtin_prefetch(Msk + (b * S_ + q0 + half * 8) * S_ + key0 + 16 + l15, 0, 1);
        }

        const int tsh = (kt & 3) << 3;
        #pragma unroll
        for (int r = 0; r < 8; ++r) {
            const int q = q0 + r + (half << 3);
            float s = acc[r] * 0.25f;                       // 1/sqrt(16)
            const unsigned mk = (Msk[(b * S_ + q) * S_ + key0 + l15] != 0);
            mb |= mk << (tsh + r);
            if (mk) s = -1.0e9f;
            // single-exp online update: one factor is always exp(0)
            const float d  = s - rm[r];
            const float e  = __expf(-fabsf(d));
            const bool  gt = d > 0.0f;
            rs[r] = gt ? fmaf(rs[r], e, 1.0f) : (rs[r] + e);
            rm[r] = gt ? s : rm[r];
        }
        if ((kt & 3) == 3) { mlds[wave][kt >> 2][lane] = mb; mb = 0; }
    }

    // merge (max,sum) pairs across the 16-lane group holding one row
    #pragma unroll
    for (int off = 8; off >= 1; off >>= 1) {
        #pragma unroll
        for (int r = 0; r < 8; ++r) {
            const float om = __shfl_xor(rm[r], off, 32);
            const float os = __shfl_xor(rs[r], off, 32);
            const float d  = om - rm[r];
            const float e  = __expf(-fabsf(d));
            const bool  gt = d > 0.0f;
            rs[r] = gt ? fmaf(rs[r], e, os) : fmaf(os, e, rs[r]);
            rm[r] = gt ? om : rm[r];
        }
    }
    float inv[8];
    #pragma unroll
    for (int r = 0; r < 8; ++r) inv[r] = 1.0f / rs[r];

    // ---------------- pass 2: recompute scores, write attn, accumulate context
    float* pt = &ptile[wave][0];
    float* attnOut = out + (size_t)(B_ * H_ * S_ * D_);  // context first, attn second
    v8f ctx = {};

    unsigned mbr = 0;
    for (int kt = 0; kt < NT; ++kt) {
        const int key0 = kt << 4;
        const float* krow = K + qkvBase + (key0 + l15) * D_;

        if ((kt & 3) == 0) mbr = mlds[wave][kt >> 2][lane];

        v8f acc = {};
        #pragma unroll
        for (int j = 0; j < 4; ++j) {
            v2f kb = *(const v2f*)(krow + j * 4 + koff);
            acc = __builtin_amdgcn_wmma_f32_16x16x4_f32(false, qa[j], false, kb,
                                                        (short)0, acc, false, false);
        }

        const int tsh = (kt & 3) << 3;
        #pragma unroll
        for (int r = 0; r < 8; ++r) {
            const int m = r + (half << 3);
            float s = acc[r] * 0.25f;
            if ((mbr >> (tsh + r)) & 1u) s = -1.0e9f;
            const float p = __expf(s - rm[r]) * inv[r];     // normalized attn
            pt[m * PT_STRIDE + l15] = p;                    // C-layout -> LDS tile
        }

        // context WMMA: A = normalized p tile (A-layout via LDS), B = V tile
        #pragma unroll
        for (int j = 0; j < 4; ++j) {
            const int k0 = j * 4 + koff;
            v2f pa;
            pa.x = pt[l15 * PT_STRIDE + k0];
            pa.y = pt[l15 * PT_STRIDE + k0 + 1];
            v2f vb;
            vb.x = V[qkvBase + (key0 + k0    ) * D_ + l15];
            vb.y = V[qkvBase + (key0 + k0 + 1) * D_ + l15];
            ctx = __builtin_amdgcn_wmma_f32_16x16x4_f32(false, pa, false, vb,
                                                        (short)0, ctx, false, false);
        }

        // attn writeback: read tile row-major from LDS, 2x global_store_b128/lane
        #pragma unroll
        for (int i = 0; i < 2; ++i) {
            const int f   = i * 128 + lane * 4;             // flat elem in 16x16 tile
            const int row = f >> 4;
            const int col = f & 15;
            const v4f pv = *(const v4f*)(pt + row * PT_STRIDE + col);
            *(v4f*)(attnOut + (size_t)(bh * S_ + q0 + row) * S_ + key0 + col) = pv;
        }
    }

    // ctx rows are already normalized (p included 1/rowsum)
    #pragma unroll
    for (int r = 0; r < 8; ++r) {
        const int q = q0 + r + (half << 3);
        out[(bh * S_ + q) * D_ + l15] = ctx[r];
    }
}

extern "C" void kernel_launch(void* const* d_in, const int* in_sizes, int n_in,
                              void* d_out, int out_size, void* d_ws, size_t ws_size,
                              hipStream_t stream) {
    (void)in_sizes; (void)n_in; (void)d_ws; (void)ws_size; (void)out_size;
    const float* Q  = (const float*)d_in[0];
    const float* K  = (const float*)d_in[1];
    const float* V  = (const float*)d_in[2];
    const int*   Mk = (const int*)d_in[3];
    float* out = (float*)d_out;

    dim3 grid((B_ * H_ * (S_ / 16)) / WAVES_PER_BLOCK);  // 512 blocks
    dim3 block(WAVES_PER_BLOCK * 32);                    // 256 threads = 8 waves
    hipLaunchKernelGGL(sdpa_wmma_f32_kernel, grid, block, 0, stream, Q, K, V, Mk, out);
}